// GCN_H_47218870452448
// MI455X (gfx1250) — compile-verified
//
#include <hip/hip_runtime.h>
#include <math.h>

// ---------------- static problem sizes (from the reference) ----------------
#define F_IN  500
#define NHID  256
#define N0    1024
#define KEEP1 768   // ceil(0.75*1024)
#define KEEP2 576   // ceil(0.75*768)
#define KEEP3 432   // ceil(0.75*576)

typedef __attribute__((ext_vector_type(2))) float v2f;
typedef __attribute__((ext_vector_type(8))) float v8f;

// ===========================================================================
// FP32 WMMA GEMM: C[M,N=256] = A[M,K] @ B[K,256]   (row-major)
// lda = K, ldb = ldc = NHID, tilesN = 16 -- ALL compile-time so addressing is
// 32-bit base + immediate offsets and the K-loop unrolls into dense WMMA runs.
// One wave per 16x16 tile; grids sized exactly (tiles % 4 waves == 0) so EXEC
// is all-ones for every V_WMMA_F32_16X16X4_F32 (ISA requirement).
// A 16x4 layout: lanes 0-15 hold M=lane, VGPR{0,1}=K{0,1}; lanes 16-31 K{2,3}.
// ===========================================================================
template<int K>
__global__ void k_gemm_f32(const float* __restrict__ A,
                           const float* __restrict__ B,
                           float* __restrict__ C)
{
  const int lane = threadIdx.x & 31;
  const int wave = threadIdx.x >> 5;
  const int tile = blockIdx.x * 4 + wave;
  const int tm   = tile >> 4;               // tilesN = 256/16 = 16
  const int tn   = tile & 15;
  const int m    = tm * 16 + (lane & 15);
  const int n    = tn * 16 + (lane & 15);
  const int kh   = (lane >> 4) * 2;         // K-pair owned by this half-wave

  const float* Ap = A + m * K + kh;         // contiguous float2 per step
  const float* Bp = B + kh * NHID + n;      // two b32 rows per step

  v8f acc = {};
  #pragma unroll 4
  for (int k = 0; k < K; k += 4) {
    v2f a = *(const v2f*)(Ap + k);
    v2f b;
    b.x = Bp[k * NHID];
    b.y = Bp[k * NHID + NHID];
    acc = __builtin_amdgcn_wmma_f32_16x16x4_f32(false, a, false, b,
                                                (short)0, acc, false, false);
  }
  const int row0 = tm * 16 + (lane >> 4) * 8;
  float* Cp = C + row0 * NHID + n;
  #pragma unroll
  for (int r = 0; r < 8; ++r)
    Cp[r * NHID] = acc[r];
}

// Gram matrix G = X @ X^T (1024x1024, K=500): B = X^T makes BOTH operands
// contiguous float2 loads (global_load_b64) -- pure WMMA + b64 inner loop.
__global__ void k_gram(const float* __restrict__ X, float* __restrict__ G)
{
  const int lane = threadIdx.x & 31;
  const int wave = threadIdx.x >> 5;
  const int tile = blockIdx.x * 4 + wave;    // 4096 tiles -> 1024 blocks
  const int tm = tile >> 6, tn = tile & 63;
  const int m  = tm * 16 + (lane & 15);
  const int n  = tn * 16 + (lane & 15);
  const int kh = (lane >> 4) * 2;

  const float* Ar = X + m * F_IN + kh;
  const float* Br = X + n * F_IN + kh;

  v8f acc = {};
  #pragma unroll 4
  for (int k = 0; k < F_IN; k += 4) {
    v2f a = *(const v2f*)(Ar + k);
    v2f b = *(const v2f*)(Br + k);
    acc = __builtin_amdgcn_wmma_f32_16x16x4_f32(false, a, false, b,
                                                (short)0, acc, false, false);
  }
  const int row0 = tm * 16 + (lane >> 4) * 8;
  float* Gp = G + row0 * N0 + n;
  #pragma unroll
  for (int r = 0; r < 8; ++r)
    Gp[r * N0] = acc[r];
}

// ---------------------------- small kernels --------------------------------
__global__ void k_init(float* out, unsigned* gammaU, int* perm0)
{
  int t = blockIdx.x * blockDim.x + threadIdx.x;
  if (t < 2 * NHID) out[t] = 0.0f;
  if (t == 0) *gammaU = 0u;            // gamma > 0, so +0.0 bits is a safe floor
  if (t < N0) perm0[t] = t;            // level-1 perm = identity
}

__global__ void k_sqnorm(const float* __restrict__ X, float* __restrict__ sq)
{
  __shared__ float red[256];
  const int node = blockIdx.x;
  float s = 0.0f;
  for (int f = threadIdx.x; f < F_IN; f += 256) {
    float v = X[node * F_IN + f];
    s += v * v;
  }
  red[threadIdx.x] = s; __syncthreads();
  for (int o = 128; o > 0; o >>= 1) {
    if (threadIdx.x < o) red[threadIdx.x] += red[threadIdx.x + o];
    __syncthreads();
  }
  if (threadIdx.x == 0) sq[node] = red[0];
}

// gamma = max_{i<j} (sq[i]+sq[j]-2G[i,j]); monotone uint trick (values >= 0)
__global__ void k_gamma(const float* __restrict__ G, const float* __restrict__ sq,
                        unsigned* gammaU)
{
  __shared__ float red[256];
  float best = 0.0f;
  for (int e = blockIdx.x * 256 + threadIdx.x; e < N0 * N0; e += 1024 * 256) {
    int i = e >> 10, j = e & (N0 - 1);
    if (i < j) best = fmaxf(best, sq[i] + sq[j] - 2.0f * G[e]);
  }
  red[threadIdx.x] = best; __syncthreads();
  for (int o = 128; o > 0; o >>= 1) {
    if (threadIdx.x < o) red[threadIdx.x] = fmaxf(red[threadIdx.x], red[threadIdx.x + o]);
    __syncthreads();
  }
  if (threadIdx.x == 0) atomicMax(gammaU, __float_as_uint(red[0]));
}

// Edge (a->b) at this level exists iff perm[a] < perm[b] && D[pa,pb] < gamma.
__global__ void k_deg(const float* __restrict__ G, const float* __restrict__ sq,
                      const unsigned* gammaU, const int* __restrict__ perm,
                      float* __restrict__ dinv, int n)
{
  int b = blockIdx.x * blockDim.x + threadIdx.x;
  if (b >= n) return;
  const float gamma = __uint_as_float(*gammaU);
  const int pb = perm[b];
  const float sqb = sq[pb];
  int cnt = 0;
  for (int a = 0; a < n; ++a) {
    int pa = perm[a];
    if (pa < pb)
      cnt += (sq[pa] + sqb - 2.0f * G[pa * N0 + pb]) < gamma;
  }
  dinv[b] = rsqrtf(1.0f + (float)cnt);
}

// M[b][a] = dinv[a]*dinv[b] if edge(a->b) else 0  -> fed to WMMA GEMM
__global__ void k_buildM(const float* __restrict__ G, const float* __restrict__ sq,
                         const unsigned* gammaU, const int* __restrict__ perm,
                         const float* __restrict__ dinv, float* __restrict__ M, int n)
{
  int idx = blockIdx.x * blockDim.x + threadIdx.x;
  if (idx >= n * n) return;
  int b = idx / n, a = idx - b * n;
  const float gamma = __uint_as_float(*gammaU);
  int pa = perm[a], pb = perm[b];
  float v = 0.0f;
  if (pa < pb) {
    float d = sq[pa] + sq[pb] - 2.0f * G[pa * N0 + pb];
    if (d < gamma) v = dinv[a] * dinv[b];
  }
  M[idx] = v;
}

// x = relu(agg + h*dinv^2 + bias)
__global__ void k_gcn_epi(const float* __restrict__ agg, const float* __restrict__ h,
                          const float* __restrict__ dinv, const float* __restrict__ bias,
                          float* __restrict__ xc, int n)
{
  int idx = blockIdx.x * blockDim.x + threadIdx.x;
  if (idx >= n * NHID) return;
  int b = idx >> 8, f = idx & (NHID - 1);
  float di = dinv[b];
  xc[idx] = fmaxf(agg[idx] + h[idx] * di * di + bias[f], 0.0f);
}

// srel[b] = x[b]·Wrel ; sroot[b] = x[b]·Wroot
__global__ void k_dots(const float* __restrict__ xc, const float* __restrict__ Wrel,
                       const float* __restrict__ Wroot,
                       float* __restrict__ srel, float* __restrict__ sroot, int n)
{
  int b = blockIdx.x * blockDim.x + threadIdx.x;
  if (b >= n) return;
  float s1 = 0.0f, s2 = 0.0f;
  for (int f = 0; f < NHID; ++f) {
    float v = xc[b * NHID + f];
    s1 += v * Wrel[f];
    s2 += v * Wroot[f];
  }
  srel[b] = s1; sroot[b] = s2;
}

// score[b] = sum_{edges a->b} srel[a] + brel + sroot[b]
__global__ void k_score(const float* __restrict__ G, const float* __restrict__ sq,
                        const unsigned* gammaU, const int* __restrict__ perm,
                        const float* __restrict__ srel, const float* __restrict__ sroot,
                        const float* __restrict__ brel, float* __restrict__ score, int n)
{
  int b = blockIdx.x * blockDim.x + threadIdx.x;
  if (b >= n) return;
  const float gamma = __uint_as_float(*gammaU);
  const int pb = perm[b];
  const float sqb = sq[pb];
  float s = 0.0f;
  for (int a = 0; a < n; ++a) {
    int pa = perm[a];
    if (pa < pb && (sq[pa] + sqb - 2.0f * G[pa * N0 + pb]) < gamma)
      s += srel[a];
  }
  score[b] = s + brel[0] + sroot[b];
}

// Stable-descending rank == jax.lax.top_k order (ties -> lower index first)
__global__ void k_rank(const float* __restrict__ score, int* __restrict__ rank, int n)
{
  int b = blockIdx.x * blockDim.x + threadIdx.x;
  if (b >= n) return;
  float sb = score[b];
  int r = 0;
  for (int a = 0; a < n; ++a) {
    float sa = score[a];
    r += (sa > sb) || (sa == sb && a < b);
  }
  rank[b] = r;
}

// xpool[rank] = x * tanh(score); permNext[rank] = perm  (for rank < k)
__global__ void k_gather(const float* __restrict__ xc, const float* __restrict__ score,
                         const int* __restrict__ rank, const int* __restrict__ permCur,
                         int* __restrict__ permNext, float* __restrict__ xpool,
                         int n, int kk)
{
  int idx = blockIdx.x * blockDim.x + threadIdx.x;
  if (idx >= n * NHID) return;
  int b = idx >> 8, f = idx & (NHID - 1);
  int r = rank[b];
  if (r < kk) {
    xpool[r * NHID + f] = xc[idx] * tanhf(score[b]);
    if (f == 0) permNext[r] = permCur[b];
  }
}

// out[f] += max_r xpool[r][f] ; out[256+f] += mean_r xpool[r][f]
__global__ void k_readout(const float* __restrict__ xp, int kk, float* __restrict__ out)
{
  int f = threadIdx.x;                 // 256 threads
  float mx = -INFINITY, sm = 0.0f;
  for (int r = 0; r < kk; ++r) {
    float v = xp[r * NHID + f];
    mx = fmaxf(mx, v);
    sm += v;
  }
  out[f]        += mx;
  out[NHID + f] += sm / (float)kk;
}

// ---------------------------------------------------------------------------
struct Ws {
  float *G, *sq, *dinv, *srel, *sroot, *score;
  unsigned* gammaU;
  int *permA, *permB, *rank;
  float *h, *agg, *xc, *poolA, *poolB, *M;
};

template<int KIN, int N, int KK>
static void run_level(const float* xin,
                      const float* W, const float* bias,
                      const float* Wrel, const float* brel, const float* Wroot,
                      const int* permCur, int* permNext, float* xpoolNext,
                      const Ws& w, float* out, hipStream_t s)
{
  // h = xin @ W : tiles = (N/16)*16 = N, blocks = N/4 (N in {1024,768,576})
  k_gemm_f32<KIN><<<N / 4, 128, 0, s>>>(xin, W, w.h);
  k_deg   <<<(N + 255) / 256, 256, 0, s>>>(w.G, w.sq, w.gammaU, permCur, w.dinv, N);
  k_buildM<<<(N * N + 255) / 256, 256, 0, s>>>(w.G, w.sq, w.gammaU, permCur, w.dinv, w.M, N);
  // agg = M @ h : masked neighborhood aggregation as dense WMMA GEMM (K = N)
  k_gemm_f32<N><<<N / 4, 128, 0, s>>>(w.M, w.h, w.agg);
  k_gcn_epi<<<N, 256, 0, s>>>(w.agg, w.h, w.dinv, bias, w.xc, N);
  k_dots  <<<(N + 255) / 256, 256, 0, s>>>(w.xc, Wrel, Wroot, w.srel, w.sroot, N);
  k_score <<<(N + 255) / 256, 256, 0, s>>>(w.G, w.sq, w.gammaU, permCur, w.srel, w.sroot, brel, w.score, N);
  k_rank  <<<(N + 255) / 256, 256, 0, s>>>(w.score, w.rank, N);
  k_gather<<<N, 256, 0, s>>>(w.xc, w.score, w.rank, permCur, permNext, xpoolNext, N, KK);
  k_readout<<<1, 256, 0, s>>>(xpoolNext, KK, out);
}

extern "C" void kernel_launch(void* const* d_in, const int* in_sizes, int n_in,
                              void* d_out, int out_size, void* d_ws, size_t ws_size,
                              hipStream_t stream)
{
  (void)in_sizes; (void)n_in; (void)out_size; (void)ws_size;
  const float* feature = (const float*)d_in[0];
  const float* W1 = (const float*)d_in[1];  const float* b1 = (const float*)d_in[2];
  const float* W2 = (const float*)d_in[3];  const float* b2 = (const float*)d_in[4];
  const float* W3 = (const float*)d_in[5];  const float* b3 = (const float*)d_in[6];
  const float* p1Wrel = (const float*)d_in[7];  const float* p1brel = (const float*)d_in[8];
  const float* p1Wroot = (const float*)d_in[9];
  const float* p2Wrel = (const float*)d_in[10]; const float* p2brel = (const float*)d_in[11];
  const float* p2Wroot = (const float*)d_in[12];
  const float* p3Wrel = (const float*)d_in[13]; const float* p3brel = (const float*)d_in[14];
  const float* p3Wroot = (const float*)d_in[15];
  float* out = (float*)d_out;

  // --- workspace layout (floats), ~13.7 MB total ---
  float* ws = (float*)d_ws;
  size_t off = 0;
  Ws w;
  w.G      = ws + off; off += (size_t)N0 * N0;
  w.sq     = ws + off; off += N0;
  w.dinv   = ws + off; off += N0;
  w.srel   = ws + off; off += N0;
  w.sroot  = ws + off; off += N0;
  w.score  = ws + off; off += N0;
  w.gammaU = (unsigned*)(ws + off); off += N0;      // padded slot
  w.permA  = (int*)(ws + off); off += N0;
  w.permB  = (int*)(ws + off); off += N0;
  w.rank   = (int*)(ws + off); off += N0;
  w.h      = ws + off; off += (size_t)N0 * NHID;
  w.agg    = ws + off; off += (size_t)N0 * NHID;
  w.xc     = ws + off; off += (size_t)N0 * NHID;
  w.poolA  = ws + off; off += (size_t)N0 * NHID;
  w.poolB  = ws + off; off += (size_t)N0 * NHID;
  w.M      = ws + off; off += (size_t)N0 * N0;

  // --- graph construction: Gram matrix (WMMA), norms, gamma ---
  k_init  <<<4, 256, 0, stream>>>(out, w.gammaU, w.permA);
  k_sqnorm<<<N0, 256, 0, stream>>>(feature, w.sq);
  k_gram  <<<1024, 128, 0, stream>>>(feature, w.G);   // 4096 tiles / 4 waves
  k_gamma <<<1024, 256, 0, stream>>>(w.G, w.sq, w.gammaU);

  // --- level 1: 1024 nodes, F_IN=500 -> keep 768 ---
  run_level<F_IN, N0, KEEP1>(feature, W1, b1, p1Wrel, p1brel, p1Wroot,
                             w.permA, w.permB, w.poolA, w, out, stream);
  // --- level 2: 768 nodes -> keep 576 ---
  run_level<NHID, KEEP1, KEEP2>(w.poolA, W2, b2, p2Wrel, p2brel, p2Wroot,
                                w.permB, w.permA, w.poolB, w, out, stream);
  // --- level 3: 576 nodes -> keep 432 ---
  run_level<NHID, KEEP2, KEEP3>(w.poolB, W3, b3, p3Wrel, p3brel, p3Wroot,
                                w.permA, w.permB, w.poolA, w, out, stream);
}